// loss_yolov1_88235808129428
// MI455X (gfx1250) — compile-verified
//
#include <hip/hip_runtime.h>

typedef __attribute__((ext_vector_type(2))) float v2f;
typedef __attribute__((ext_vector_type(4))) float v4f;
typedef __attribute__((ext_vector_type(8))) float v8f;

#define YOLO_S       7
#define YOLO_BATCH   32768
#define YOLO_CELLS   (YOLO_BATCH * YOLO_S * YOLO_S)  /* 1,605,632 cells, 12 ch each */
#define MAIN_BLOCK   256
#define MAIN_GRID    2048

// ---------------------------------------------------------------------------
// Wave32 sum using the matrix pipe: one V_WMMA_F32_16X16X4_F32 with A = ones
// collapses x_n + x_{n+16} (B rows 0/1 = the two lane-halves of B's VGPR0,
// rows 2/3 = 0), then 4 shuffle-xor steps finish the 16 column sums.
// Must be called with EXEC all-1s (full block, after reconvergence).
// Fixed dataflow => bitwise deterministic.
// ---------------------------------------------------------------------------
__device__ __forceinline__ float wave_sum32(float x) {
#if __has_builtin(__builtin_amdgcn_wmma_f32_16x16x4_f32)
    v2f a; a[0] = 1.0f; a[1] = 1.0f;   // A[m][k] = 1 for all 16x4 slots
    v2f b; b[0] = x;    b[1] = 0.0f;   // B row0 = lanes 0-15, row1 = lanes 16-31
    v8f c = {};
    v8f d = __builtin_amdgcn_wmma_f32_16x16x4_f32(
        /*neg_a=*/false, a, /*neg_b=*/false, b,
        /*c_mod=*/(short)0, c, /*reuse_a=*/false, /*reuse_b=*/false);
    float y = d[0];                    // lane L: x_{L&15} + x_{(L&15)+16}
    y += __shfl_xor(y, 8, 32);
    y += __shfl_xor(y, 4, 32);
    y += __shfl_xor(y, 2, 32);
    y += __shfl_xor(y, 1, 32);
    return y;
#else
    for (int off = 16; off > 0; off >>= 1) x += __shfl_xor(x, off, 32);
    return x;
#endif
}

// Block-level reduce of 3 accumulators; returns sums in thread 0.
__device__ __forceinline__ void block_sum3(float& A, float& B, float& C) {
    A = wave_sum32(A);
    B = wave_sum32(B);
    C = wave_sum32(C);
    __shared__ float sA[8], sB[8], sC[8];
    const int lane = threadIdx.x & 31;
    const int wid  = threadIdx.x >> 5;
    if (lane == 0) { sA[wid] = A; sB[wid] = B; sC[wid] = C; }
    __syncthreads();
    if (threadIdx.x == 0) {
        float tA = 0.0f, tB = 0.0f, tC = 0.0f;
        const int nw = (int)(blockDim.x >> 5);
        for (int w = 0; w < nw; ++w) { tA += sA[w]; tB += sB[w]; tC += sC[w]; }
        A = tA; B = tB; C = tC;
    }
}

// ---------------------------------------------------------------------------
// Main streaming kernel: grid-stride over cells; 3x global_load_b128 per
// tensor per cell; per-thread partials; WMMA-assisted wave/block reduce;
// deterministic per-block partial writes into ws (SoA: [3][nblocks]).
// ---------------------------------------------------------------------------
__global__ __launch_bounds__(MAIN_BLOCK) void yolo_loss_main(
    const float* __restrict__ pred, const float* __restrict__ label,
    float* __restrict__ partials, int nblocks) {
    const int tid    = blockIdx.x * blockDim.x + threadIdx.x;
    const int stride = gridDim.x * blockDim.x;
    const v4f* __restrict__ p4 = (const v4f*)pred;
    const v4f* __restrict__ l4 = (const v4f*)label;

    float accA = 0.0f;   // loss_conf + 5*(xy+wh) + cls (obj cells)
    float accB = 0.0f;   // noobj squared conf error
    float accC = 0.0f;   // noobj cell count

    const float inv_s = 1.0f / (float)YOLO_S;

    for (int i = tid; i < YOLO_CELLS; i += stride) {
        const size_t base = (size_t)i * 3;          // 12 floats = 3 x float4
        v4f p0 = p4[base + 0], p1 = p4[base + 1], p2 = p4[base + 2];
        v4f l0 = l4[base + 0], l1 = l4[base + 1], l2 = l4[base + 2];

        if (i + stride < YOLO_CELLS) {              // -> global_prefetch_b8
            __builtin_prefetch(p4 + base + (size_t)stride * 3, 0, 3);
            __builtin_prefetch(l4 + base + (size_t)stride * 3, 0, 3);
        }

        // channel map: [0..3]=box0 xywh, [4]=conf0, [5..8]=box1 xywh,
        //              [9]=conf1, [10..11]=classes
        const float pconf0 = p1[0], pconf1 = p2[1];
        const float lconf  = l1[0];                 // label[...,4] == obj flag
        const bool  obj    = lconf > 0.0f;

        if (!obj) {
            const float d4 = pconf0 - lconf;
            const float d9 = pconf1 - l2[1];
            accB += d4 * d4 + d9 * d9;
            accC += 1.0f;
        } else {
            // label box0 corners (reference computes areas from corners)
            const float lcx = l0[0] * inv_s, lcy = l0[1] * inv_s;
            const float lx0 = lcx - 0.5f * l0[2], ly0 = lcy - 0.5f * l0[3];
            const float lx1 = lcx + 0.5f * l0[2], ly1 = lcy + 0.5f * l0[3];
            const float la  = (lx1 - lx0) * (ly1 - ly0);

            // pred box0 corners
            const float c0x = p0[0] * inv_s, c0y = p0[1] * inv_s;
            const float b0x0 = c0x - 0.5f * p0[2], b0y0 = c0y - 0.5f * p0[3];
            const float b0x1 = c0x + 0.5f * p0[2], b0y1 = c0y + 0.5f * p0[3];
            // pred box1 corners (ch 5..8 = p1.yzw, p2.x)
            const float c1x = p1[1] * inv_s, c1y = p1[2] * inv_s;
            const float b1x0 = c1x - 0.5f * p1[3], b1y0 = c1y - 0.5f * p2[0];
            const float b1x1 = c1x + 0.5f * p1[3], b1y1 = c1y + 0.5f * p2[0];

            // IoU of each pred box with label box0
            float iw, ih, inter, a1;
            iw = fmaxf(fminf(b0x1, lx1) - fmaxf(b0x0, lx0), 0.0f);
            ih = fmaxf(fminf(b0y1, ly1) - fmaxf(b0y0, ly0), 0.0f);
            inter = iw * ih;
            a1 = (b0x1 - b0x0) * (b0y1 - b0y0);
            const float iou0 = inter / (a1 + la - inter);

            iw = fmaxf(fminf(b1x1, lx1) - fmaxf(b1x0, lx0), 0.0f);
            ih = fmaxf(fminf(b1y1, ly1) - fmaxf(b1y0, ly0), 0.0f);
            inter = iw * ih;
            a1 = (b1x1 - b1x0) * (b1y1 - b1y0);
            const float iou1 = inter / (a1 + la - inter);

            const bool  sel1 = iou1 > iou0;          // jnp.argmax first-max
            const float miou = fmaxf(iou0, iou1);

            // responsible pred box
            const float rpx = sel1 ? p1[1] : p0[0];
            const float rpy = sel1 ? p1[2] : p0[1];
            const float rpw = sel1 ? p1[3] : p0[2];
            const float rph = sel1 ? p2[0] : p0[3];
            const float rpc = sel1 ? pconf1 : pconf0;
            // responsible label box (box1 chans 5..8 = l1.yzw, l2.x)
            const float rlx = sel1 ? l1[1] : l0[0];
            const float rly = sel1 ? l1[2] : l0[1];
            const float rlw = sel1 ? l1[3] : l0[2];
            const float rlh = sel1 ? l2[0] : l0[3];

            const float dcf = rpc - miou;
            const float dx = rpx - rlx, dy = rpy - rly;
            const float dw = rpw - rlw, dh = rph - rlh;
            const float dk0 = p2[2] - l2[2], dk1 = p2[3] - l2[3];

            accA += dcf * dcf
                  + 5.0f * (dx * dx + dy * dy + dw * dw + dh * dh)
                  + dk0 * dk0 + dk1 * dk1;
        }
    }

    block_sum3(accA, accB, accC);
    if (threadIdx.x == 0) {
        partials[              blockIdx.x] = accA;
        partials[    nblocks + blockIdx.x] = accB;
        partials[2 * nblocks + blockIdx.x] = accC;
    }
}

// ---------------------------------------------------------------------------
// Single-block deterministic final reduce + loss combine.
// ---------------------------------------------------------------------------
__global__ __launch_bounds__(MAIN_BLOCK) void yolo_loss_reduce(
    const float* __restrict__ partials, int nblocks, float* __restrict__ out) {
    float a = 0.0f, b = 0.0f, c = 0.0f;
    for (int j = threadIdx.x; j < nblocks; j += (int)blockDim.x) {
        a += partials[j];
        b += partials[nblocks + j];
        c += partials[2 * nblocks + j];
    }
    block_sum3(a, b, c);
    if (threadIdx.x == 0) {
        // loss = (0.1 * sum_sq / (2*count) + obj_terms) / BATCH
        out[0] = (0.1f * b / (2.0f * c) + a) * (1.0f / (float)YOLO_BATCH);
    }
}

extern "C" void kernel_launch(void* const* d_in, const int* in_sizes, int n_in,
                              void* d_out, int out_size, void* d_ws, size_t ws_size,
                              hipStream_t stream) {
    (void)in_sizes; (void)n_in; (void)out_size;
    const float* pred  = (const float*)d_in[0];
    const float* label = (const float*)d_in[1];
    float* partials = (float*)d_ws;   // [3][nblocks] floats
    float* out      = (float*)d_out;

    // Clamp block count to available workspace (needs 12 bytes per block).
    int nblocks = MAIN_GRID;
    const size_t cap = ws_size / (3 * sizeof(float));
    if ((size_t)nblocks > cap) nblocks = (int)(cap ? cap : 1);

    yolo_loss_main<<<nblocks, MAIN_BLOCK, 0, stream>>>(pred, label, partials, nblocks);
    yolo_loss_reduce<<<1, MAIN_BLOCK, 0, stream>>>(partials, nblocks, out);
}